// NNUE_5832565588369
// MI455X (gfx1250) — compile-verified
//
#include <hip/hip_runtime.h>

// NNUE evaluation for MI455X (gfx1250).
// Bandwidth-bound: 1.34 GB of streamed activations, ~2.75 GFLOP -> AI ~2 FLOP/B.
// Strategy: f32 WMMA (16x16x4) for the feature transformer, non-temporal loads
// for the streamed matrices, L2-cached ft_w, f32 atomics for k-split reduction,
// tiny tail kernel for bias/stm-mix/clip/l1/l2.
//
// Round-1 fix: the ft_w (B-operand) load is now unconditional from row (n & 3)
// with a value-level select to zero for N>=4 -> no exec-mask divergence in the
// inner loop (previous version emitted s_and_saveexec/s_cbranch per K-step).

#define NNUE_F     40960   // feature count (compile-time for address strength reduction)
#define FCHUNKS    8       // split of feature axis across blockIdx.y
#define WAVES      4       // waves per workgroup, further k-split
#define TILE_ROWS  16      // batch rows per workgroup (WMMA M)

typedef float v2f __attribute__((ext_vector_type(2)));
typedef float v8f __attribute__((ext_vector_type(8)));

// ---------------------------------------------------------------------------
// Kernel 0: zero the partial-sum workspace (B x 8 floats)
// ---------------------------------------------------------------------------
__global__ void nnue_zero_ws(float* __restrict__ ws, int n) {
    int i = blockIdx.x * blockDim.x + threadIdx.x;
    if (i < n) ws[i] = 0.0f;
}

// ---------------------------------------------------------------------------
// Kernel 1: feature transformer via V_WMMA_F32_16X16X4_F32.
//   grid.x = B/16 row tiles, grid.y = FCHUNKS feature chunks, 4 waves/WG.
//   Each wave owns a private K-range and accumulates a 16x16 f32 tile for
//   wfts and one for bfts (only N=0..3 columns meaningful).
//   Partials reduced into ws[row*8 + {0..3 | 4..7}] with f32 global atomics.
// ---------------------------------------------------------------------------
__global__ __launch_bounds__(WAVES * 32)
void nnue_ft_wmma(const float* __restrict__ wfts,
                  const float* __restrict__ bfts,
                  const float* __restrict__ ftw,   // [4, NNUE_F] row-major
                  float* __restrict__ ws)          // [B, 8]
{
    const int lane = threadIdx.x & 31;
    const int wave = threadIdx.x >> 5;
    const int n    = lane & 15;   // WMMA N index (output column) / A row index
    const int half = lane >> 4;   // lane group: K offset 0 or 2

    const int rowbase = blockIdx.x * TILE_ROWS;
    const int row     = rowbase + n;

    const size_t F = (size_t)NNUE_F;
    const float* __restrict__ wrow = wfts + (size_t)row * F;
    const float* __restrict__ brow = bfts + (size_t)row * F;
    // All lanes load a VALID ft_w row: lanes 4..15 mirror rows 0..3 (same
    // cache lines -> WGP$ hits, no divergence). Zeroing happens by select.
    const float* __restrict__ ftwn = ftw + (size_t)(n & 3) * F;
    const bool bactive = (n < 4);   // loop-invariant -> lives in VCC

    const int chunk = NNUE_F / FCHUNKS;   // 5120
    const int sub   = chunk / WAVES;      // 1280
    const int k0    = blockIdx.y * chunk + wave * sub;
    const int kend  = k0 + sub;
    const int koff  = half * 2;           // this lane's K offset inside a K=4 step

    v8f acc_w = {0.f, 0.f, 0.f, 0.f, 0.f, 0.f, 0.f, 0.f};
    v8f acc_b = {0.f, 0.f, 0.f, 0.f, 0.f, 0.f, 0.f, 0.f};

    for (int k = k0; k < kend; k += 16) {
        // Prefetch the stream ~2KB ahead (speculative).
        __builtin_prefetch(wrow + k + 512, 0, 1);
        __builtin_prefetch(brow + k + 512, 0, 1);
#pragma unroll
        for (int u = 0; u < 16; u += 4) {
            const int kk = k + u + koff;
            // A tiles: lane (L&15) = batch row, VGPR pair = K koff..koff+1.
            // Streamed once -> non-temporal so ft_w keeps the caches.
            v2f aw = __builtin_nontemporal_load((const v2f*)(wrow + kk));
            v2f ab = __builtin_nontemporal_load((const v2f*)(brow + kk));
            // B tile: column n of ft_w for the same K pair (regular/cached
            // load, unconditional), zeroed by v_cndmask for n >= 4.
            v2f braw = *(const v2f*)(ftwn + kk);
            v2f bw;
            bw.x = bactive ? braw.x : 0.0f;
            bw.y = bactive ? braw.y : 0.0f;
            // D = A x B + C   (exact f32 semantics, 16x16x4)
            acc_w = __builtin_amdgcn_wmma_f32_16x16x4_f32(
                false, aw, false, bw, (short)0, acc_w, false, false);
            acc_b = __builtin_amdgcn_wmma_f32_16x16x4_f32(
                false, ab, false, bw, (short)0, acc_b, false, false);
        }
    }

    // C/D layout: VGPR j on lane L holds element (M = j + 8*(L>>4), N = L&15).
    if (bactive) {
#pragma unroll
        for (int j = 0; j < 8; ++j) {
            const int orow = rowbase + j + 8 * half;
            atomicAdd(&ws[orow * 8 + n],     acc_w[j]);
            atomicAdd(&ws[orow * 8 + 4 + n], acc_b[j]);
        }
    }
}

// ---------------------------------------------------------------------------
// Kernel 2: per-row tail. acc = stm*[w,b] + (1-stm)*[b,w]; clip; 8x8 layer;
// clip; 8->1 layer.
// ---------------------------------------------------------------------------
__global__ void nnue_tail(const float* __restrict__ ws,
                          const float* __restrict__ stm,
                          const float* __restrict__ ftb,
                          const float* __restrict__ l1w,  // [8,8]
                          const float* __restrict__ l1b,  // [8]
                          const float* __restrict__ l2w,  // [1,8]
                          const float* __restrict__ l2b,  // [1]
                          float* __restrict__ out, int B)
{
    int row = blockIdx.x * blockDim.x + threadIdx.x;
    if (row >= B) return;

    float w[4], b[4];
#pragma unroll
    for (int i = 0; i < 4; ++i) {
        w[i] = ws[row * 8 + i]     + ftb[i];
        b[i] = ws[row * 8 + 4 + i] + ftb[i];
    }

    const float s  = stm[row];
    const float os = 1.0f - s;

    float x[8];
#pragma unroll
    for (int i = 0; i < 4; ++i) {
        x[i]     = fminf(fmaxf(s * w[i] + os * b[i], 0.0f), 1.0f);
        x[4 + i] = fminf(fmaxf(s * b[i] + os * w[i], 0.0f), 1.0f);
    }

    float y[8];
#pragma unroll
    for (int o = 0; o < 8; ++o) {
        float acc = l1b[o];
#pragma unroll
        for (int i = 0; i < 8; ++i) acc = fmaf(x[i], l1w[o * 8 + i], acc);
        y[o] = fminf(fmaxf(acc, 0.0f), 1.0f);
    }

    float r = l2b[0];
#pragma unroll
    for (int i = 0; i < 8; ++i) r = fmaf(y[i], l2w[i], r);
    out[row] = r;
}

// ---------------------------------------------------------------------------
extern "C" void kernel_launch(void* const* d_in, const int* in_sizes, int n_in,
                              void* d_out, int out_size, void* d_ws, size_t ws_size,
                              hipStream_t stream)
{
    const float* wfts = (const float*)d_in[0];
    const float* bfts = (const float*)d_in[1];
    const float* stm  = (const float*)d_in[2];
    const float* ftw  = (const float*)d_in[3];
    const float* ftb  = (const float*)d_in[4];
    const float* l1w  = (const float*)d_in[5];
    const float* l1b  = (const float*)d_in[6];
    const float* l2w  = (const float*)d_in[7];
    const float* l2b  = (const float*)d_in[8];

    const int B = in_sizes[2];          // stm has one element per batch row
    float* ws = (float*)d_ws;           // B x 8 f32 partial sums

    const int wsn = B * 8;
    nnue_zero_ws<<<(wsn + 255) / 256, 256, 0, stream>>>(ws, wsn);

    dim3 grid(B / TILE_ROWS, FCHUNKS);
    nnue_ft_wmma<<<grid, WAVES * 32, 0, stream>>>(wfts, bfts, ftw, ws);

    nnue_tail<<<(B + 255) / 256, 256, 0, stream>>>(
        ws, stm, ftb, l1w, l1b, l2w, l2b, (float*)d_out, B);
}